// SmallNet_27144193310806
// MI455X (gfx1250) — compile-verified
//
#include <hip/hip_runtime.h>
#include <math.h>

#ifndef __has_builtin
#define __has_builtin(x) 0
#endif

typedef float v2f __attribute__((ext_vector_type(2)));
typedef float v8f __attribute__((ext_vector_type(8)));
typedef int   v4i __attribute__((ext_vector_type(4)));

typedef __attribute__((address_space(1))) v4i as1_v4i;   // global-memory v4i
typedef __attribute__((address_space(3))) v4i as3_v4i;   // LDS v4i

#define NPTS      4096
#define BETA_C    1.0f
#define SQRT_PI_C 1.77245385090551602729f

// ---------------------------------------------------------------------------
// CDNA5 async global->LDS staging (ASYNCcnt path). The builtin exists on this
// toolchain (per round-1 diagnostic) and takes v4i pointers.
// ---------------------------------------------------------------------------
__device__ __forceinline__ void cdna5_async_copy_b128(void* lds_dst, const void* g_src) {
#if __has_builtin(__builtin_amdgcn_global_load_async_to_lds_b128)
  __builtin_amdgcn_global_load_async_to_lds_b128(
      (as1_v4i*)(unsigned long long)g_src,
      (as3_v4i*)(unsigned int)(unsigned long long)lds_dst,
      0, 0);
#else
  *(float4*)lds_dst = *(const float4*)g_src;
#endif
}

__device__ __forceinline__ void cdna5_async_wait() {
#if __has_builtin(__builtin_amdgcn_global_load_async_to_lds_b128)
#if __has_builtin(__builtin_amdgcn_s_wait_asynccnt)
  __builtin_amdgcn_s_wait_asynccnt(0);
#else
  asm volatile("s_wait_asynccnt 0" ::: "memory");
#endif
#endif
}

// ---------------------------------------------------------------------------
// Wave32 sum via V_WMMA_F32_16X16X4_F32 with an all-ones B matrix.
// Each lane's partial occupies exactly one A slot (a[1]=0), so column n of
// D = A*ones is the total sum, split across lanes n and n+16 of C/D.
// Returns the full-wave sum in every lane. EXEC must be all ones (call from
// uniform control flow with a full block).
// ---------------------------------------------------------------------------
__device__ __forceinline__ float cdna5_wave_sum(float p) {
  v2f a; a[0] = p;    a[1] = 0.0f;
  v2f b; b[0] = 1.0f; b[1] = 1.0f;
  v8f c = {0.0f, 0.0f, 0.0f, 0.0f, 0.0f, 0.0f, 0.0f, 0.0f};
  c = __builtin_amdgcn_wmma_f32_16x16x4_f32(false, a, false, b, (short)0, c, false, false);
  float s = ((c[0] + c[1]) + (c[2] + c[3])) + ((c[4] + c[5]) + (c[6] + c[7]));
  s += __shfl_xor(s, 16, 32);   // pair lane n with n+16 -> full column sum
  return s;
}

// ---------------------------------------------------------------------------
__global__ void zero_out_kernel(float* out, int n) {
  int i = blockIdx.x * blockDim.x + threadIdx.x;
  if (i < n) out[i] = 0.0f;
}

// Event term: sum_e [ beta - || (z0_u - z0_v) + (v0_u - v0_v) * t_e ||^2 ]
// z0/v0 (64 KB total) staged in LDS; gathers served by ds_load-class ops.
__global__ __launch_bounds__(256) void event_kernel(
    const float* __restrict__ z0, const float* __restrict__ v0,
    const float* __restrict__ et, const int* __restrict__ ui,
    const int* __restrict__ vi, float* __restrict__ out, int n_events) {
  __shared__ __align__(16) float s_z[NPTS * 2];
  __shared__ __align__(16) float s_v[NPTS * 2];
  const int tid = threadIdx.x;

  // Stage 2048 float4 per array via async b128 copies (8 rounds of 256 lanes).
  for (int q = tid; q < (NPTS * 2) / 4; q += 256) {
    cdna5_async_copy_b128(&((float4*)s_z)[q], &((const float4*)z0)[q]);
    cdna5_async_copy_b128(&((float4*)s_v)[q], &((const float4*)v0)[q]);
  }
  cdna5_async_wait();
  __syncthreads();

  float acc = 0.0f;
  const int stride = gridDim.x * 256;
  for (int e = blockIdx.x * 256 + tid; e < n_events; e += stride) {
    float t = et[e];
    int u = ui[e], v = vi[e];
    float dzx = s_z[2 * u]     - s_z[2 * v];
    float dzy = s_z[2 * u + 1] - s_z[2 * v + 1];
    float dvx = s_v[2 * u]     - s_v[2 * v];
    float dvy = s_v[2 * u + 1] - s_v[2 * v + 1];
    float dx = fmaf(dvx, t, dzx);
    float dy = fmaf(dvy, t, dzy);
    acc += BETA_C - fmaf(dx, dx, dy * dy);
  }

  float w = cdna5_wave_sum(acc);
  if ((tid & 31) == 0) atomicAdd(out, w);
}

// Non-event term: integrand is symmetric in (i,j), so compute the full
// off-diagonal sum and scale by 0.5. Thread owns point i, loops an LDS-staged
// 256-point j chunk. alpha == beta == 1 (reference constants).
__global__ __launch_bounds__(256) void pair_kernel(
    const float* __restrict__ z0, const float* __restrict__ v0,
    const float* __restrict__ t0p, const float* __restrict__ tnp,
    float* __restrict__ out) {
  __shared__ __align__(16) float s_zj[256 * 2];
  __shared__ __align__(16) float s_vj[256 * 2];
  const int tid = threadIdx.x;
  const int i   = blockIdx.x * 256 + tid;
  const int j0  = blockIdx.y * 256;

  // Stage the j-chunk: 128 float4 per array; one b128 async op per lane.
  if (tid < 128) {
    cdna5_async_copy_b128(&((float4*)s_zj)[tid],       &((const float4*)(z0 + 2 * j0))[tid]);
  } else {
    cdna5_async_copy_b128(&((float4*)s_vj)[tid - 128], &((const float4*)(v0 + 2 * j0))[tid - 128]);
  }
  cdna5_async_wait();
  __syncthreads();

  const float t0 = t0p[0];
  const float tn = tnp[0];
  const float xi = z0[2 * i], yi = z0[2 * i + 1];
  const float mi = v0[2 * i], ni = v0[2 * i + 1];

  float acc = 0.0f;
  for (int jj = 0; jj < 256; ++jj) {
    int j = j0 + jj;
    float a = xi - s_zj[2 * jj];
    float b = yi - s_zj[2 * jj + 1];
    float m = mi - s_vj[2 * jj];
    float n = ni - s_vj[2 * jj + 1];
    float mn2 = fmaf(m, m, n * n);
    float rs  = rsqrtf(mn2);                 // alpha==1: s = sqrt(mn2), 1/s = rs
    float cross = fmaf(a, n, -b * m);
    float dot   = fmaf(a, m,  b * n);
    float cr    = cross * rs;
    float expo  = expf(BETA_C - cr * cr);    // (-cross^2 + mn2*beta)/mn2
    float e0    = erff(fmaf(mn2, t0, dot) * rs);
    float e1    = erff(fmaf(mn2, tn, dot) * rs);
    float integ = -SQRT_PI_C * expo * (e0 - e1) * (0.5f * rs);
    acc += (j == i) ? 0.0f : integ;          // select, keeps NaN from j==i out
  }
  acc *= -0.5f;   // triu sum = half the off-diagonal sum; objective subtracts it

  float w = cdna5_wave_sum(acc);
  if ((tid & 31) == 0) atomicAdd(out, w);
}

// ---------------------------------------------------------------------------
extern "C" void kernel_launch(void* const* d_in, const int* in_sizes, int n_in,
                              void* d_out, int out_size, void* d_ws, size_t ws_size,
                              hipStream_t stream) {
  const float* z0 = (const float*)d_in[0];
  const float* v0 = (const float*)d_in[1];
  const float* et = (const float*)d_in[2];
  const float* t0 = (const float*)d_in[3];
  const float* tn = (const float*)d_in[4];
  const int*   ui = (const int*)d_in[5];
  const int*   vi = (const int*)d_in[6];
  float* out = (float*)d_out;

  const int n_events = in_sizes[2];
  const int npts     = in_sizes[0] / 2;   // 4096

  zero_out_kernel<<<1, 32, 0, stream>>>(out, out_size);
  event_kernel<<<512, 256, 0, stream>>>(z0, v0, et, ui, vi, out, n_events);

  dim3 grid(npts / 256, npts / 256);      // 16 x 16 blocks, 256 j's per block
  pair_kernel<<<grid, 256, 0, stream>>>(z0, v0, t0, tn, out);
}